// PPRPowerIteration_17428977287556
// MI455X (gfx1250) — compile-verified
//
#include <hip/hip_runtime.h>
#include <hip/hip_bf16.h>

#define N_NODES   100000
#define N_EDGES   3200000
#define IN_FEATS  512
#define N_HIDDEN  64
#define N_CLASSES 64
#define ALPHA_F   0.1f
#define NITER     10

#define WAVES_PER_BLOCK 8
#define N_TILES (N_NODES / 16)   // 6250 exactly

typedef __attribute__((ext_vector_type(2))) float v2f;
typedef __attribute__((ext_vector_type(8))) float v8f;

// ---------------------------------------------------------------------------
// Fused MLP: local = tanh(x @ w1) @ w2, one wave per 16-row tile.
// Layer-1 WMMA f32 16x16x4 over K=512; h staged in LDS (C-layout -> A-layout
// reshuffle); layer-2 WMMA over K=64 with w2 resident in LDS.
// ---------------------------------------------------------------------------
__global__ __launch_bounds__(256) void mlp_wmma_kernel(
    const float* __restrict__ x, const float* __restrict__ w1,
    const float* __restrict__ w2, float* __restrict__ local) {
  __shared__ float w2s[N_HIDDEN * N_CLASSES];                  // 16 KB
  __shared__ float hs[WAVES_PER_BLOCK][16 * N_HIDDEN];         // 32 KB

  for (int i = threadIdx.x; i < N_HIDDEN * N_CLASSES; i += 256) w2s[i] = w2[i];
  __syncthreads();

  const int wave = threadIdx.x >> 5;
  const int lane = threadIdx.x & 31;
  const int half = lane >> 4;      // 0: lanes 0-15, 1: lanes 16-31
  const int lm   = lane & 15;
  const int tile = blockIdx.x * WAVES_PER_BLOCK + wave;
  if (tile >= N_TILES) return;     // wave-uniform: EXEC stays all-ones for WMMA
  const int r0 = tile * 16;

  // ---------------- layer 1: h = tanh(x @ w1) ----------------
  v8f a0 = {0,0,0,0,0,0,0,0}, a1 = {0,0,0,0,0,0,0,0};
  v8f a2 = {0,0,0,0,0,0,0,0}, a3 = {0,0,0,0,0,0,0,0};
  // A-frag: lane lm holds row r0+lm; VGPR0/1 = K = k0+2*half, k0+2*half+1
  const float* arow = x + (size_t)(r0 + lm) * IN_FEATS + 2 * half;
  for (int k0 = 0; k0 < IN_FEATS; k0 += 4) {
    v2f a = *(const v2f*)(arow + k0);                      // 8B-aligned b64 load
    // B-frag: VGPR0 = row (k0+2*half), VGPR1 = row (k0+2*half+1), col = nt*16+lm
    const float* bp = w1 + (size_t)(k0 + 2 * half) * N_HIDDEN + lm;
    v2f b0 = { bp[ 0], bp[N_HIDDEN +  0] };
    v2f b1 = { bp[16], bp[N_HIDDEN + 16] };
    v2f b2 = { bp[32], bp[N_HIDDEN + 32] };
    v2f b3 = { bp[48], bp[N_HIDDEN + 48] };
    a0 = __builtin_amdgcn_wmma_f32_16x16x4_f32(false, a, false, b0, (short)0, a0, false, false);
    a1 = __builtin_amdgcn_wmma_f32_16x16x4_f32(false, a, false, b1, (short)0, a1, false, false);
    a2 = __builtin_amdgcn_wmma_f32_16x16x4_f32(false, a, false, b2, (short)0, a2, false, false);
    a3 = __builtin_amdgcn_wmma_f32_16x16x4_f32(false, a, false, b3, (short)0, a3, false, false);
  }

  // tanh + stage h tile to LDS. C layout: lane<16 -> M=v, N=lane; lane>=16 -> M=v+8.
  float* hrow = &hs[wave][0];
#pragma unroll
  for (int v = 0; v < 8; ++v) {
    const int m = v + 8 * half;
    hrow[m * N_HIDDEN + 0 * 16 + lm] = tanhf(a0[v]);
    hrow[m * N_HIDDEN + 1 * 16 + lm] = tanhf(a1[v]);
    hrow[m * N_HIDDEN + 2 * 16 + lm] = tanhf(a2[v]);
    hrow[m * N_HIDDEN + 3 * 16 + lm] = tanhf(a3[v]);
  }
  // Same-wave DS RAW: DS is in-order per wave; explicit wait as belt-and-suspenders.
  asm volatile("s_wait_dscnt 0" ::: "memory");

  // ---------------- layer 2: local = h @ w2 ----------------
  v8f d0 = {0,0,0,0,0,0,0,0}, d1 = {0,0,0,0,0,0,0,0};
  v8f d2 = {0,0,0,0,0,0,0,0}, d3 = {0,0,0,0,0,0,0,0};
  for (int k0 = 0; k0 < N_HIDDEN; k0 += 4) {
    v2f a = *(const v2f*)(hrow + lm * N_HIDDEN + k0 + 2 * half);   // ds_load_b64
    const float* bp = w2s + (k0 + 2 * half) * N_CLASSES + lm;
    v2f b0 = { bp[ 0], bp[N_CLASSES +  0] };
    v2f b1 = { bp[16], bp[N_CLASSES + 16] };
    v2f b2 = { bp[32], bp[N_CLASSES + 32] };
    v2f b3 = { bp[48], bp[N_CLASSES + 48] };
    d0 = __builtin_amdgcn_wmma_f32_16x16x4_f32(false, a, false, b0, (short)0, d0, false, false);
    d1 = __builtin_amdgcn_wmma_f32_16x16x4_f32(false, a, false, b1, (short)0, d1, false, false);
    d2 = __builtin_amdgcn_wmma_f32_16x16x4_f32(false, a, false, b2, (short)0, d2, false, false);
    d3 = __builtin_amdgcn_wmma_f32_16x16x4_f32(false, a, false, b3, (short)0, d3, false, false);
  }

#pragma unroll
  for (int v = 0; v < 8; ++v) {
    const size_t grow = (size_t)(r0 + v + 8 * half) * N_CLASSES;
    local[grow + 0 * 16 + lm] = d0[v];
    local[grow + 1 * 16 + lm] = d1[v];
    local[grow + 2 * 16 + lm] = d2[v];
    local[grow + 3 * 16 + lm] = d3[v];
  }
}

// ---------------------------------------------------------------------------
// next = ALPHA * local  (element-wise; safe when next == local)
// ---------------------------------------------------------------------------
__global__ __launch_bounds__(256) void alpha_init_kernel(
    const float* __restrict__ local, float* __restrict__ next, int n4) {
  int i = blockIdx.x * 256 + threadIdx.x;
  if (i >= n4) return;
  float4 l = ((const float4*)local)[i];
  float4 o;
  o.x = ALPHA_F * l.x; o.y = ALPHA_F * l.y;
  o.z = ALPHA_F * l.z; o.w = ALPHA_F * l.w;
  ((float4*)next)[i] = o;
}

// ---------------------------------------------------------------------------
// Edge scatter: one wave per edge, 2 channels per lane.
// next[row,:] += val * cur[col,:]   (L2-resident gather + fp32 HW atomics)
// ---------------------------------------------------------------------------
__global__ __launch_bounds__(256) void scatter_kernel(
    const int* __restrict__ erow, const int* __restrict__ ecol,
    const float* __restrict__ eval, const float* __restrict__ cur,
    float* __restrict__ next) {
  const int gtid = blockIdx.x * 256 + threadIdx.x;
  const int e    = gtid >> 5;
  if (e >= N_EDGES) return;
  const int lane = gtid & 31;
  const int r = erow[e];
  const int c = ecol[e];
  const float v = eval[e];
  const int ch = lane * 2;
  const float2 p = *(const float2*)(cur + (size_t)c * N_CLASSES + ch);
  float* dst = next + (size_t)r * N_CLASSES + ch;
  unsafeAtomicAdd(dst + 0, v * p.x);   // global_atomic_add_f32
  unsafeAtomicAdd(dst + 1, v * p.y);
}

// ---------------------------------------------------------------------------
extern "C" void kernel_launch(void* const* d_in, const int* in_sizes, int n_in,
                              void* d_out, int out_size, void* d_ws, size_t ws_size,
                              hipStream_t stream) {
  const float* x    = (const float*)d_in[0];
  const float* w1   = (const float*)d_in[1];
  const float* w2   = (const float*)d_in[2];
  const int*   erow = (const int*)d_in[3];
  const int*   ecol = (const int*)d_in[4];
  const float* eval = (const float*)d_in[5];
  float* out = (float*)d_out;

  float* P0 = (float*)d_ws;
  float* P1 = P0 + (size_t)N_NODES * N_CLASSES;

  // local_preds -> d_out; d_out doubles as preds_0 and as alpha-source until iter 9.
  mlp_wmma_kernel<<<(N_TILES + WAVES_PER_BLOCK - 1) / WAVES_PER_BLOCK, 256, 0, stream>>>(
      x, w1, w2, out);

  const int n4 = N_NODES * N_CLASSES / 4;
  const int init_blocks = (n4 + 255) / 256;
  const int scat_blocks = N_EDGES / 8;   // 32 threads per edge, 256/block

  const float* cur = out;
  for (int it = 0; it < NITER; ++it) {
    float* next = (it == NITER - 1) ? out : ((it & 1) ? P1 : P0);
    alpha_init_kernel<<<init_blocks, 256, 0, stream>>>(out, next, n4);
    scatter_kernel<<<scat_blocks, 256, 0, stream>>>(erow, ecol, eval, cur, next);
    cur = next;
  }
  // final result already in d_out (iteration 9 targeted it directly)
}